// ComposeTransform_75608604279136
// MI455X (gfx1250) — compile-verified
//
#include <hip/hip_runtime.h>

// ComposeTransform: out = u2 + trilinear(u1, grid + u2), B=2, D=H=W=160, C=3, f32.
// Memory-bound (~295 MB HBM traffic -> ~12.7us at 23.3 TB/s). Strategy:
//  - warp2: streamed once, staged per-wave via async global->LDS b128 (CDNA5 ASYNCcnt path)
//  - warp1: cached gathers; 3D-slab block order keeps working set in 192MB L2
//  - out:   non-temporal stores (don't pollute L2 needed for warp1 reuse)

#ifndef __has_builtin
#define __has_builtin(x) 0
#endif

#if __has_builtin(__builtin_amdgcn_global_load_async_to_lds_b128) && \
    __has_builtin(__builtin_amdgcn_s_wait_asynccnt)
#define CT_USE_ASYNC 1
#else
#define CT_USE_ASYNC 0
#endif

namespace {
constexpr int kB = 2;
constexpr int kD = 160;
constexpr int kH = 160;
constexpr int kW = 160;
constexpr int kC = 3;
constexpr int kSH = kW * kC;        // 480  elements per h-row
constexpr int kSD = kH * kW * kC;   // 76800 elements per d-plane
constexpr long long kVol = (long long)kD * kH * kW * kC;

typedef int v4i_ct __attribute__((vector_size(4 * sizeof(int))));
typedef v4i_ct __attribute__((address_space(1))) gv4i_ct;  // global
typedef v4i_ct __attribute__((address_space(3))) lv4i_ct;  // LDS

__device__ __forceinline__ int imin_i(int a, int b) { return a < b ? a : b; }
}  // namespace

__global__ __launch_bounds__(256)
void ComposeTransform_75608604279136_kernel(const float* __restrict__ warp1,
                                            const float* __restrict__ warp2,
                                            float* __restrict__ out) {
  // Block tile: 32(w) x 4(h) x 2(d). Grid: (W/32, H/4, (D/2)*B).
  const int w  = blockIdx.x * 32 + threadIdx.x;
  const int h  = blockIdx.y * 4  + threadIdx.y;
  const int zb = blockIdx.z;
  const int d  = (zb % (kD / 2)) * 2 + threadIdx.z;
  const int b  = zb / (kD / 2);

  const int  vox = (d * kH + h) * kW + w;        // < 160^3, fits int32
  const long long p = (long long)b * kVol + (long long)vox * kC;

  float u2x, u2y, u2z;

#if CT_USE_ASYNC
  // Each wave == one (d,h) row of 32 voxels == 384 contiguous, 16B-aligned bytes
  // (base byte offset is 12 * 32k, a multiple of 16). Stage via async global->LDS
  // (24 lanes x b128), wait on ASYNCcnt. No barrier needed: every wave consumes
  // only the row it loaded itself.
  __shared__ alignas(16) float s2[8 * 96];       // 8 rows * 32 voxels * 3ch
  const int row = threadIdx.z * 4 + threadIdx.y; // == wave id in block
  {
    const float* grow = warp2 + (long long)b * kVol +
                        (long long)((d * kH + h) * kW + blockIdx.x * 32) * kC;
    if (threadIdx.x < 24) {
      __builtin_amdgcn_global_load_async_to_lds_b128(
          (gv4i_ct*)(grow + threadIdx.x * 4),
          (lv4i_ct*)(&s2[row * 96 + threadIdx.x * 4]),
          0, 0);
    }
    __builtin_amdgcn_s_wait_asynccnt(0);
  }
  u2x = s2[row * 96 + threadIdx.x * 3 + 0];
  u2y = s2[row * 96 + threadIdx.x * 3 + 1];
  u2z = s2[row * 96 + threadIdx.x * 3 + 2];
#else
  {
    const float* p2 = warp2 + p;
    u2x = __builtin_nontemporal_load(p2 + 0);
    u2y = __builtin_nontemporal_load(p2 + 1);
    u2z = __builtin_nontemporal_load(p2 + 2);
  }
#endif

  // Sample location = grid + u2, border-clipped (matches jnp.clip semantics).
  float cx = fminf(fmaxf((float)d + u2x, 0.0f), (float)(kD - 1));
  float cy = fminf(fmaxf((float)h + u2y, 0.0f), (float)(kH - 1));
  float cz = fminf(fmaxf((float)w + u2z, 0.0f), (float)(kW - 1));

  const float fx = floorf(cx), fy = floorf(cy), fz = floorf(cz);
  const int ix0 = (int)fx, iy0 = (int)fy, iz0 = (int)fz;
  const float wx = cx - fx, wy = cy - fy, wz = cz - fz;
  const int ix1 = imin_i(ix0 + 1, kD - 1);
  const int iy1 = imin_i(iy0 + 1, kH - 1);
  const int iz1 = imin_i(iz0 + 1, kW - 1);

  const float* __restrict__ w1b = warp1 + (long long)b * kVol;
  const int r00 = ix0 * kSD + iy0 * kSH;
  const int r01 = ix0 * kSD + iy1 * kSH;
  const int r10 = ix1 * kSD + iy0 * kSH;
  const int r11 = ix1 * kSD + iy1 * kSH;
  const int z0 = iz0 * kC, z1 = iz1 * kC;

  // Warm the far-plane cachelines while we compute weights (global_prefetch_b8).
  __builtin_prefetch(w1b + r10 + z0, 0, 3);
  __builtin_prefetch(w1b + r11 + z0, 0, 3);

  const float wx0 = 1.0f - wx, wy0 = 1.0f - wy, wz0 = 1.0f - wz;

  float a0 = u2x, a1 = u2y, a2 = u2z;  // out = u2 + sum(corner * weight)
#define CT_ACC(OFF, WGT)                          \
  do {                                            \
    const float* q = w1b + (OFF);                 \
    const float ww = (WGT);                       \
    a0 += q[0] * ww; a1 += q[1] * ww; a2 += q[2] * ww; \
  } while (0)

  CT_ACC(r00 + z0, wx0 * wy0 * wz0);
  CT_ACC(r00 + z1, wx0 * wy0 * wz);
  CT_ACC(r01 + z0, wx0 * wy  * wz0);
  CT_ACC(r01 + z1, wx0 * wy  * wz);
  CT_ACC(r10 + z0, wx  * wy0 * wz0);
  CT_ACC(r10 + z1, wx  * wy0 * wz);
  CT_ACC(r11 + z0, wx  * wy  * wz0);
  CT_ACC(r11 + z1, wx  * wy  * wz);
#undef CT_ACC

  float* po = out + p;
  __builtin_nontemporal_store(a0, po + 0);
  __builtin_nontemporal_store(a1, po + 1);
  __builtin_nontemporal_store(a2, po + 2);
}

extern "C" void kernel_launch(void* const* d_in, const int* in_sizes, int n_in,
                              void* d_out, int out_size, void* d_ws, size_t ws_size,
                              hipStream_t stream) {
  (void)in_sizes; (void)n_in; (void)out_size; (void)d_ws; (void)ws_size;
  const float* warp1 = (const float*)d_in[0];
  const float* warp2 = (const float*)d_in[1];
  float* out = (float*)d_out;

  dim3 block(32, 4, 2);
  dim3 grid(kW / 32, kH / 4, (kD / 2) * kB);
  ComposeTransform_75608604279136_kernel<<<grid, block, 0, stream>>>(warp1, warp2, out);
}